// BeeAlgorithm_50964081934826
// MI455X (gfx1250) — compile-verified
//
#include <hip/hip_runtime.h>
#include <hip/hip_bf16.h>
#include <stdint.h>

// Bee algorithm step on MI455X (gfx1250, wave32).
// Memory-bound (~250MB streamed @ 23.3TB/s). Row norms computed as the
// diagonal of X·X^T via V_WMMA_F32_16X16X4_F32 (16 rows per wave, tile staged
// in LDS, fed as both A and transposed B). Noise is a counter-based RNG so the
// onlooker population is never materialized; only elite positions (16MB) live
// in the workspace and the winning row is regenerated at the end.
//
// Round-2 fix: wave_row_norms16 is force-inlined so address-space inference
// sees the __shared__ tile and emits ds_* (not flat_*) for LDS traffic,
// decoupling WMMA operand waits (DScnt) from the global stream (LOADcnt).

#define S_N 2048
#define E_N 512
#define O_N 4096
#define D_N 8192
#define RADIUS 0.1f
#define BATCH_N 8
#define TILE_K 64
#define LDS_PITCH 68   // 68 floats = 272B: keeps float4 slots 16B aligned

#define SEED_SCOUT 0xA341u
#define SEED_ELITE 0xB712u
#define SEED_SEL   0xC923u
#define SEED_ONL   0xD534u

typedef __attribute__((ext_vector_type(2))) float v2f;
typedef __attribute__((ext_vector_type(8))) float v8f;

__device__ __forceinline__ uint32_t pcg_hash(uint32_t v) {
    v = v * 747796405u + 2891336453u;
    uint32_t w = ((v >> ((v >> 28u) + 4u)) ^ v) * 277803737u;
    return (w >> 22u) ^ w;
}

__device__ __forceinline__ float gauss_noise(uint32_t seed, uint32_t idx) {
    uint32_t h1 = pcg_hash(idx ^ (seed * 0x9E3779B9u));
    uint32_t h2 = pcg_hash(h1 + 0x85EBCA6Bu);
    float u1 = ((float)((h1 >> 8) + 1u)) * (1.0f / 16777217.0f);  // (0,1]
    float u2 = ((float)(h2 >> 8)) * (1.0f / 16777216.0f);
    return __builtin_sqrtf(-2.0f * __logf(u1)) * __cosf(6.28318530718f * u2);
}

// One wave computes L2 norms of 16 rows: rows row_base..row_base+15 of the
// logical population; source row in `base` is row_map[r] if row_map!=null.
// value(r,d) = base[src(r)*D+d] + nscale*N(0,1) keyed on (seed, r*D+d).
// Accumulates the 16x16 Gram matrix with V_WMMA_F32_16X16X4_F32; diagonal
// element m lives at (VGPR m%8, lane m<8 ? m : m+16) per the CDNA5 C/D layout.
// Writes sqrt of the diagonal to out16[0..15].
// Force-inlined: the LDS pointer must resolve to addrspace(3) at each call
// site so tile traffic lowers to ds_* instead of flat_*.
__device__ __attribute__((always_inline)) __forceinline__
void wave_row_norms16(const float* __restrict__ base,
                      const int* __restrict__ row_map,
                      int row_base, uint32_t seed, float nscale,
                      float* __restrict__ lds, float* __restrict__ out16) {
    const int lane = threadIdx.x & 31;
    const int half = lane >> 4;     // 0 or 1
    const int l16  = lane & 15;

    // Source row for each of this lane's 8 tile-load rows (rows 2i+half).
    int rsrc[8];
#pragma unroll
    for (int i = 0; i < 8; ++i) {
        int rlog = row_base + 2 * i + half;
        rsrc[i] = row_map ? row_map[rlog] : rlog;
    }

    v8f acc = {};
    for (int k0 = 0; k0 < D_N; k0 += TILE_K) {
        // Stage a perturbed 16 x TILE_K tile into LDS. Each half-wave reads
        // 256B contiguous per row (16 lanes x float4), 2 rows per pass.
#pragma unroll
        for (int i = 0; i < 8; ++i) {
            int r = 2 * i + half;
            int rlog = row_base + r;
            int col = l16 * 4;
            const float* p = base + (size_t)rsrc[i] * D_N + k0 + col;
            __builtin_prefetch(p + TILE_K, 0, 1);   // global_prefetch_b8
            float4 v = *(const float4*)p;
            uint32_t nidx = (uint32_t)rlog * D_N + (uint32_t)(k0 + col);
            v.x += nscale * gauss_noise(seed, nidx + 0u);
            v.y += nscale * gauss_noise(seed, nidx + 1u);
            v.z += nscale * gauss_noise(seed, nidx + 2u);
            v.w += nscale * gauss_noise(seed, nidx + 3u);
            *(float4*)&lds[r * LDS_PITCH + col] = v;
        }
        __builtin_amdgcn_wave_barrier();

        // Gram accumulate: D += A x B, A = tile (16x4 chunk), B = tile^T.
        // A layout: VGPR0 = K {0|2}, VGPR1 = K {1|3} by half-wave; M = lane%16.
        // B layout: VGPR0 = row K {0|1}, VGPR1 = row K {2|3}; N = lane%16.
#pragma unroll
        for (int kk = 0; kk < TILE_K; kk += 4) {
            v2f a, b;
            a.x = lds[l16 * LDS_PITCH + kk + 2 * half + 0];
            a.y = lds[l16 * LDS_PITCH + kk + 2 * half + 1];
            b.x = lds[l16 * LDS_PITCH + kk + half + 0];
            b.y = lds[l16 * LDS_PITCH + kk + half + 2];
            acc = __builtin_amdgcn_wmma_f32_16x16x4_f32(
                false, a, false, b, (short)0, acc, false, false);
        }
        __builtin_amdgcn_wave_barrier();
    }

    // Extract diagonal: m in [0,8) at lane m (acc[m]); m in [8,16) at lane
    // 16+m i.e. lanes 24..31 (acc[m-8]).
#pragma unroll
    for (int m = 0; m < 8; ++m) {
        if (lane == m)      out16[m]     = __builtin_sqrtf(acc[m]);
        if (lane == 24 + m) out16[8 + m] = __builtin_sqrtf(acc[m]);
    }
    __builtin_amdgcn_wave_barrier();
}

// ---- Kernel 1: scout fitness (perturbation regenerated, not stored) --------
__global__ void k_scout_fit(const float* __restrict__ scout_pos,
                            float* __restrict__ scout_fit) {
    __shared__ float lds[8][16 * LDS_PITCH];
    int wave = threadIdx.x >> 5;
    int row_base = (blockIdx.x * 8 + wave) * 16;
    wave_row_norms16(scout_pos, nullptr, row_base, SEED_SCOUT, RADIUS,
                     lds[wave], scout_fit + row_base);
}

// ---- Kernel 2: full bitonic sort of 2048 fitnesses, keep best 512 ----------
__global__ void k_topk(const float* __restrict__ fit,
                       int* __restrict__ eidx, float* __restrict__ efit) {
    __shared__ float key[S_N];
    __shared__ int   val[S_N];
    int tid = threadIdx.x;                       // 1024 threads
    for (int i = tid; i < S_N; i += 1024) { key[i] = fit[i]; val[i] = i; }
    __syncthreads();
    for (int k = 2; k <= S_N; k <<= 1) {
        for (int j = k >> 1; j > 0; j >>= 1) {
            for (int t = tid; t < S_N; t += 1024) {
                int ixj = t ^ j;
                if (ixj > t) {
                    bool up = ((t & k) == 0);
                    float a = key[t], b = key[ixj];
                    if (up ? (a > b) : (a < b)) {
                        key[t] = b; key[ixj] = a;
                        int va = val[t]; val[t] = val[ixj]; val[ixj] = va;
                    }
                }
            }
            __syncthreads();
        }
    }
    if (tid < E_N) { efit[tid] = key[tid]; eidx[tid] = val[tid]; }
}

// ---- Kernel 3: gather elite positions (regenerate scout perturbation) ------
__global__ void k_elite_gather(const float* __restrict__ scout_pos,
                               const int* __restrict__ eidx,
                               float* __restrict__ elite_pos) {
    int t = blockIdx.x * blockDim.x + threadIdx.x;   // one float4 each
    int e = t / (D_N / 4);
    int d = (t % (D_N / 4)) * 4;
    int s = eidx[e];
    float4 v = *(const float4*)(scout_pos + (size_t)s * D_N + d);
    uint32_t nidx = (uint32_t)s * D_N + (uint32_t)d;
    v.x += RADIUS * gauss_noise(SEED_SCOUT, nidx + 0u);
    v.y += RADIUS * gauss_noise(SEED_SCOUT, nidx + 1u);
    v.z += RADIUS * gauss_noise(SEED_SCOUT, nidx + 2u);
    v.w += RADIUS * gauss_noise(SEED_SCOUT, nidx + 3u);
    *(float4*)(elite_pos + (size_t)e * D_N + d) = v;
}

// ---- Kernel 4: one greedy elite step (fused candidate norm + accept) -------
__global__ void k_elite_step(float* __restrict__ elite_pos,
                             float* __restrict__ elite_fit, uint32_t seed) {
    __shared__ float lds[8][16 * LDS_PITCH];
    __shared__ float cand[8][16];
    int wave = threadIdx.x >> 5;
    int lane = threadIdx.x & 31;
    int row_base = (blockIdx.x * 8 + wave) * 16;
    wave_row_norms16(elite_pos, nullptr, row_base, seed, RADIUS * 0.5f,
                     lds[wave], cand[wave]);
    __builtin_amdgcn_wave_barrier();
    int half = lane >> 4, l16 = lane & 15;
    for (int i = 0; i < 8; ++i) {
        int r = 2 * i + half;
        int e = row_base + r;
        if (cand[wave][r] < elite_fit[e]) {   // accepted: apply the noise step
            for (int k = l16 * 4; k < D_N; k += 64) {
                float4 v = *(const float4*)(elite_pos + (size_t)e * D_N + k);
                uint32_t nidx = (uint32_t)e * D_N + (uint32_t)k;
                v.x += RADIUS * 0.5f * gauss_noise(seed, nidx + 0u);
                v.y += RADIUS * 0.5f * gauss_noise(seed, nidx + 1u);
                v.z += RADIUS * 0.5f * gauss_noise(seed, nidx + 2u);
                v.w += RADIUS * 0.5f * gauss_noise(seed, nidx + 3u);
                *(float4*)(elite_pos + (size_t)e * D_N + k) = v;
            }
        }
    }
    __builtin_amdgcn_wave_barrier();
    if (lane < 16) {
        int e = row_base + lane;
        float cf = cand[wave][lane];
        if (cf < elite_fit[e]) elite_fit[e] = cf;
    }
}

// ---- Kernel 5: softmax(-fit) categorical selection of 4096 onlookers -------
__global__ void k_select(const float* __restrict__ efit,
                         int* __restrict__ sel, uint32_t seed) {
    __shared__ float cdf[E_N];
    __shared__ float red[E_N];
    int tid = threadIdx.x;                      // 512 threads
    float logit = -efit[tid];
    red[tid] = logit;
    __syncthreads();
    for (int off = 256; off > 0; off >>= 1) {
        if (tid < off) red[tid] = fmaxf(red[tid], red[tid + off]);
        __syncthreads();
    }
    float m = red[0];
    __syncthreads();
    cdf[tid] = __expf(logit - m);
    __syncthreads();
    for (int off = 1; off < E_N; off <<= 1) {   // Hillis-Steele inclusive scan
        float add = (tid >= off) ? cdf[tid - off] : 0.0f;
        __syncthreads();
        cdf[tid] += add;
        __syncthreads();
    }
    float total = cdf[E_N - 1];
    for (int o = tid; o < O_N; o += 512) {
        uint32_t h = pcg_hash((uint32_t)o ^ (seed * 0x9E3779B9u));
        float u = ((float)(h >> 8)) * (1.0f / 16777216.0f) * total;
        int lo = 0, hi = E_N - 1;
        while (lo < hi) {
            int mid = (lo + hi) >> 1;
            if (cdf[mid] < u) lo = mid + 1; else hi = mid;
        }
        sel[o] = lo;
    }
}

// ---- Kernel 6: onlooker candidate fitness + accept flag --------------------
__global__ void k_onl_fit(const float* __restrict__ elite_pos,
                          const float* __restrict__ elite_fit,
                          const int* __restrict__ sel,
                          float* __restrict__ onl_fit,
                          int* __restrict__ accept, uint32_t seed) {
    __shared__ float lds[8][16 * LDS_PITCH];
    __shared__ float cand[8][16];
    int wave = threadIdx.x >> 5;
    int lane = threadIdx.x & 31;
    int row_base = (blockIdx.x * 8 + wave) * 16;
    wave_row_norms16(elite_pos, sel, row_base, seed, RADIUS * 0.3f,
                     lds[wave], cand[wave]);
    __builtin_amdgcn_wave_barrier();
    if (lane < 16) {
        int o = row_base + lane;
        float cf = cand[wave][lane];
        onl_fit[o] = cf;                              // stored regardless
        accept[o] = (cf < elite_fit[sel[o]]) ? 1 : 0; // base fit = elite norm
    }
}

// ---- Kernel 7: global argmin over 6656 fitnesses + improved flag -----------
__global__ void k_argmin(const float* __restrict__ sf, const float* __restrict__ ef,
                         const float* __restrict__ of,
                         const float* __restrict__ best_fit_in,
                         int* __restrict__ winner) {
    __shared__ float bv[1024];
    __shared__ int   bi[1024];
    int tid = threadIdx.x;
    float best = 3.0e38f; int idx = 0x7fffffff;
    for (int i = tid; i < S_N + E_N + O_N; i += 1024) {
        float f = (i < S_N) ? sf[i]
                : (i < S_N + E_N) ? ef[i - S_N]
                : of[i - S_N - E_N];
        if (f < best || (f == best && i < idx)) { best = f; idx = i; }
    }
    bv[tid] = best; bi[tid] = idx;
    __syncthreads();
    for (int off = 512; off > 0; off >>= 1) {
        if (tid < off) {
            if (bv[tid + off] < bv[tid] ||
                (bv[tid + off] == bv[tid] && bi[tid + off] < bi[tid])) {
                bv[tid] = bv[tid + off]; bi[tid] = bi[tid + off];
            }
        }
        __syncthreads();
    }
    if (tid == 0) {
        winner[0] = (bv[0] < best_fit_in[0]) ? 1 : 0;
        winner[1] = bi[0];
    }
}

// ---- Kernel 8: regenerate winning row, broadcast to (BATCH, D) output ------
__global__ void k_emit(const float* __restrict__ scout_pos,
                       const float* __restrict__ elite_pos,
                       const int* __restrict__ sel,
                       const int* __restrict__ accept,
                       const float* __restrict__ best_position,
                       const int* __restrict__ winner,
                       float* __restrict__ out) {
    int d = blockIdx.x * blockDim.x + threadIdx.x;   // 0..D_N-1
    float v;
    if (!winner[0]) {
        v = best_position[d];
    } else {
        int i = winner[1];
        if (i < S_N) {                                // perturbed scout
            uint32_t nidx = (uint32_t)i * D_N + (uint32_t)d;
            v = scout_pos[(size_t)i * D_N + d] + RADIUS * gauss_noise(SEED_SCOUT, nidx);
        } else if (i < S_N + E_N) {                   // final elite
            v = elite_pos[(size_t)(i - S_N) * D_N + d];
        } else {                                      // onlooker
            int o = i - S_N - E_N;
            int e = sel[o];
            v = elite_pos[(size_t)e * D_N + d];
            if (accept[o]) {
                uint32_t nidx = (uint32_t)o * D_N + (uint32_t)d;
                v += RADIUS * 0.3f * gauss_noise(SEED_ONL, nidx);
            }
        }
    }
#pragma unroll
    for (int b = 0; b < BATCH_N; ++b) out[(size_t)b * D_N + d] = v;
}

extern "C" void kernel_launch(void* const* d_in, const int* in_sizes, int n_in,
                              void* d_out, int out_size, void* d_ws, size_t ws_size,
                              hipStream_t stream) {
    (void)in_sizes; (void)n_in; (void)out_size; (void)ws_size;
    const float* scout_pos = (const float*)d_in[1];
    const float* best_position = (const float*)d_in[4];
    const float* best_fitness  = (const float*)d_in[5];
    float* out = (float*)d_out;

    // Workspace layout (~16.9 MB)
    float* elite_pos = (float*)d_ws;                      // E*D
    float* scout_fit = elite_pos + (size_t)E_N * D_N;     // S
    float* elite_fit = scout_fit + S_N;                   // E
    float* onl_fit   = elite_fit + E_N;                   // O
    int*   elite_idx = (int*)(onl_fit + O_N);             // E
    int*   sel       = elite_idx + E_N;                   // O
    int*   accept    = sel + O_N;                         // O
    int*   winner    = accept + O_N;                      // 2

    k_scout_fit<<<S_N / 16 / 8, 256, 0, stream>>>(scout_pos, scout_fit);
    k_topk<<<1, 1024, 0, stream>>>(scout_fit, elite_idx, elite_fit);
    k_elite_gather<<<(E_N * (D_N / 4)) / 256, 256, 0, stream>>>(scout_pos, elite_idx, elite_pos);
    for (uint32_t t = 0; t < 3; ++t)
        k_elite_step<<<E_N / 16 / 8, 256, 0, stream>>>(elite_pos, elite_fit, SEED_ELITE + t);
    k_select<<<1, E_N, 0, stream>>>(elite_fit, sel, SEED_SEL);
    k_onl_fit<<<O_N / 16 / 8, 256, 0, stream>>>(elite_pos, elite_fit, sel,
                                                onl_fit, accept, SEED_ONL);
    k_argmin<<<1, 1024, 0, stream>>>(scout_fit, elite_fit, onl_fit, best_fitness, winner);
    k_emit<<<D_N / 256, 256, 0, stream>>>(scout_pos, elite_pos, sel, accept,
                                          best_position, winner, out);
}